// deform_up_87660282511802
// MI455X (gfx1250) — compile-verified
//
#include <hip/hip_runtime.h>
#include <math.h>
#include <stdint.h>

typedef __attribute__((ext_vector_type(2))) float v2f;
typedef __attribute__((ext_vector_type(8))) float v8f;

#define BN_EPS 1e-5f

// ---------------------------------------------------------------------------
// CDNA5 async copy global -> LDS (ASYNCcnt-tracked), inline asm per bridge doc.
// vdst VGPR holds the workgroup-relative LDS byte offset (low 32 bits of the
// generic pointer for an LDS object), vaddr is the 64-bit global address.
// ---------------------------------------------------------------------------
__device__ __forceinline__ void async_ld_b32(void* ldsPtr, const float* g) {
  uint32_t off = (uint32_t)(uintptr_t)ldsPtr;
  asm volatile("global_load_async_to_lds_b32 %0, %1, off"
               :: "v"(off), "v"(g)
               : "memory");
}
__device__ __forceinline__ void wait_async_le4() {
  asm volatile("s_wait_asynccnt 0x4" ::: "memory");
}
__device__ __forceinline__ void wait_async_le0() {
  asm volatile("s_wait_asynccnt 0x0" ::: "memory");
}

// ---------------------------------------------------------------------------
// 1) Upsample x1 (align_corners 2x trilinear) and concat with x2 -> X [128,16,24,24]
// ---------------------------------------------------------------------------
__global__ void k_upcat(const float* __restrict__ x1, const float* __restrict__ x2,
                        float* __restrict__ X) {
  const int SP = 16 * 24 * 24;
  int i = blockIdx.x * blockDim.x + threadIdx.x;
  if (i >= 128 * SP) return;
  int c = i / SP;
  int s = i - c * SP;
  if (c < 64) { X[i] = x2[c * SP + s]; return; }
  int cz = c - 64;
  int xo = s % 24;
  int yo = (s / 24) % 24;
  int zo = s / (24 * 24);
  float pz = (float)zo * (7.0f / 15.0f);
  float py = (float)yo * (11.0f / 23.0f);
  float px = (float)xo * (11.0f / 23.0f);
  int z0 = (int)floorf(pz), y0 = (int)floorf(py), x0 = (int)floorf(px);
  int z1 = min(z0 + 1, 7), y1 = min(y0 + 1, 11), x1i = min(x0 + 1, 11);
  float tz = pz - (float)z0, ty = py - (float)y0, tx = px - (float)x0;
  const float* b = x1 + cz * (8 * 12 * 12);
  float v000 = b[(z0 * 12 + y0) * 12 + x0],  v001 = b[(z0 * 12 + y0) * 12 + x1i];
  float v010 = b[(z0 * 12 + y1) * 12 + x0],  v011 = b[(z0 * 12 + y1) * 12 + x1i];
  float v100 = b[(z1 * 12 + y0) * 12 + x0],  v101 = b[(z1 * 12 + y0) * 12 + x1i];
  float v110 = b[(z1 * 12 + y1) * 12 + x0],  v111 = b[(z1 * 12 + y1) * 12 + x1i];
  float v00 = v000 * (1.f - tx) + v001 * tx;
  float v01 = v010 * (1.f - tx) + v011 * tx;
  float v10 = v100 * (1.f - tx) + v101 * tx;
  float v11 = v110 * (1.f - tx) + v111 * tx;
  float v0 = v00 * (1.f - ty) + v01 * ty;
  float v1 = v10 * (1.f - ty) + v11 * ty;
  X[i] = v0 * (1.f - tz) + v1 * tz;
}

// ---------------------------------------------------------------------------
// 2) Offset conv as implicit-im2col GEMM, async-LDS double-buffered.
//    Block = 192 threads (6 waves), one 16-column n-tile per block; wave w owns
//    m-tile w (M=81 -> 6 tiles). B chunks of K=48 staged to LDS with 4
//    global_load_async_to_lds_b32 per wave (uniform), double buffered.
//    kk = c*27 + (kz*9+ky*3+kx); WMMA f32 16x16x4, 12 per chunk.
// ---------------------------------------------------------------------------
__global__ void k_conv_gemm_wmma(const float* __restrict__ X,   // [C,D,H,W]
                                 const float* __restrict__ Wm,  // [M, C*27]
                                 const float* __restrict__ bias,// [M]
                                 float* __restrict__ OUT,       // [M, N]
                                 int C, int D, int H, int W,
                                 int M, int Do, int Ho, int Wo) {
  const int N = Do * Ho * Wo;
  const int Kdim = C * 27;
  const int nCh = Kdim / 48;           // C multiple of 16 -> exact
  __shared__ __align__(16) float Bt[2][16 * 52];   // [n][kk_local], pad stride 52

  const int nt = blockIdx.x;
  const int tid = threadIdx.x;         // 0..191
  const int lane = tid & 31;
  const int wave = tid >> 5;           // m-tile 0..5
  const int half = lane >> 4;
  const int l16 = lane & 15;

  // staging role: fixed n per thread, 12 kk-rows across threads
  const int sn = tid & 15;
  const int srow = tid >> 4;           // 0..11
  int p = nt * 16 + sn;
  int pCl = p < N ? p : N - 1;         // garbage cols never stored
  int wo = pCl % Wo;
  int tq = pCl / Wo;
  int ho = tq % Ho;
  int dz = tq / Ho;

  // compute role
  int m = wave * 16 + l16;
  int mCl = m < M ? m : M - 1;
  const float* Arow = Wm + (size_t)mCl * Kdim;

  auto stage = [&](int buf, int kk0) {
    float* dst = &Bt[buf][0];
    #pragma unroll
    for (int j = 0; j < 4; ++j) {
      int kkl = srow + 12 * j;
      int kk = kk0 + kkl;
      int c0 = kk / 27, kr = kk - c0 * 27;
      int kz = kr / 9, ky = (kr / 3) % 3, kx = kr % 3;
      const float* g = &X[((c0 * D + dz + kz) * H + ho + ky) * W + wo + kx];
      async_ld_b32(&dst[sn * 52 + kkl], g);
    }
  };

  v8f acc = {};
  stage(0, 0);
  int buf = 0;
  for (int ch = 0; ch < nCh; ++ch) {
    bool hasNext = (ch + 1) < nCh;
    if (hasNext) stage(buf ^ 1, (ch + 1) * 48);
    if (hasNext) wait_async_le4(); else wait_async_le0();
    __syncthreads();                   // current buf staged by all waves
    const float* Bb = &Bt[buf][0];
    const float* Ab = Arow + ch * 48;
    #pragma unroll
    for (int c0 = 0; c0 < 48; c0 += 4) {
      int ka = c0 + 2 * half;
      float2 aa = *(const float2*)&Ab[ka];
      float2 bb = *(const float2*)&Bb[l16 * 52 + ka];
      v2f a, b;
      a.x = aa.x; a.y = aa.y;
      b.x = bb.x; b.y = bb.y;
      acc = __builtin_amdgcn_wmma_f32_16x16x4_f32(false, a, false, b, (short)0, acc,
                                                  false, false);
    }
    __syncthreads();                   // done reading buf before it is re-staged
    buf ^= 1;
  }
  // C/D layout: VGPR r -> (M = mt*16 + r + 8*half, N = l16)
  int n = nt * 16 + l16;
  if (n < N) {
    int rbase = wave * 16 + half * 8;
    for (int r = 0; r < 8; ++r) {
      int mm = rbase + r;
      if (mm < M) OUT[mm * N + n] = acc[r] + bias[mm];
    }
  }
}

// ---------------------------------------------------------------------------
// 3a) Rearrange deform weights [64,C,27] -> [27,64,C] so A-fragments are
//     contiguous float2 loads.
// ---------------------------------------------------------------------------
__global__ void k_wt_to_kmc(const float* __restrict__ Wt, float* __restrict__ Wk,
                            int C) {
  int i = blockIdx.x * blockDim.x + threadIdx.x;
  if (i >= 64 * C * 27) return;
  int c = i % C;
  int t = i / C;
  int m = t % 64;
  int k = t / 64;
  Wk[i] = Wt[(m * C + c) * 27 + k];
}

// ---------------------------------------------------------------------------
// 3b) Deformable GEMM: OUT[o,p] = sum_{k,c} w[o,c,k] * trilerp(X[c], coord(k,p))
//     Block = 128 threads (4 waves), one 16-column n-tile; per kernel point k
//     the block gathers a [16 n x C] V-tile to LDS (double buffered, padded
//     stride 132), waves each run one 16-row m-tile through C/4 WMMA steps.
// ---------------------------------------------------------------------------
__global__ void k_deform_gemm_wmma(const float* __restrict__ X,   // [C,D,H,W]
                                   const float* __restrict__ OFF, // [81, N]
                                   const float* __restrict__ Wk,  // [27,64,C]
                                   float* __restrict__ OUT,       // [64, N]
                                   int C, int D, int H, int W,
                                   int Do, int Ho, int Wo) {
  const int N = Do * Ho * Wo;
  const int DHW = D * H * W;
  __shared__ __align__(16) float Vt[2][16 * 132];

  int nt = blockIdx.x;
  int tid = threadIdx.x;              // blockDim == 128
  int lane = tid & 31;
  int wave = tid >> 5;                // m-tile 0..3 (M = 64)
  int half = lane >> 4;
  int l16 = lane & 15;

  int gn = tid & 15;
  int gc0 = tid >> 4;                 // 0..7
  int p = nt * 16 + gn;
  bool pvalid = p < N;
  int pCl = pvalid ? p : N - 1;
  int wo = pCl % Wo;
  int tq = pCl / Wo;
  int ho = tq % Ho;
  int dz = tq / Ho;

  auto gather = [&](int buf, int k) {
    int kz = k / 9, ky = (k / 3) % 3, kx = k % 3;
    float oz = OFF[(k * 3 + 0) * N + pCl];
    float oy = OFF[(k * 3 + 1) * N + pCl];
    float ox = OFF[(k * 3 + 2) * N + pCl];
    float z = fminf(fmaxf((float)(dz + kz) + oz, 0.f), (float)(D - 1));
    float y = fminf(fmaxf((float)(ho + ky) + oy, 0.f), (float)(H - 1));
    float x = fminf(fmaxf((float)(wo + kx) + ox, 0.f), (float)(W - 1));
    int z0 = (int)floorf(z), y0 = (int)floorf(y), x0 = (int)floorf(x);
    float tz = z - (float)z0, ty = y - (float)y0, tx = x - (float)x0;
    int z1 = min(z0 + 1, D - 1), y1 = min(y0 + 1, H - 1), x1 = min(x0 + 1, W - 1);
    int s000 = (z0 * H + y0) * W + x0, s001 = (z0 * H + y0) * W + x1;
    int s010 = (z0 * H + y1) * W + x0, s011 = (z0 * H + y1) * W + x1;
    int s100 = (z1 * H + y0) * W + x0, s101 = (z1 * H + y0) * W + x1;
    int s110 = (z1 * H + y1) * W + x0, s111 = (z1 * H + y1) * W + x1;
    float uz = 1.f - tz, uy = 1.f - ty, ux = 1.f - tx;
    float w000 = uz * uy * ux, w001 = uz * uy * tx;
    float w010 = uz * ty * ux, w011 = uz * ty * tx;
    float w100 = tz * uy * ux, w101 = tz * uy * tx;
    float w110 = tz * ty * ux, w111 = tz * ty * tx;
    for (int c = gc0; c < C; c += 8) {
      const float* bp = X + c * DHW;
      float v = w000 * bp[s000] + w001 * bp[s001] + w010 * bp[s010] + w011 * bp[s011] +
                w100 * bp[s100] + w101 * bp[s101] + w110 * bp[s110] + w111 * bp[s111];
      Vt[buf][gn * 132 + c] = pvalid ? v : 0.f;
    }
  };

  v8f acc = {};
  gather(0, 0);
  __syncthreads();
  int buf = 0;
  for (int k = 0; k < 27; ++k) {
    if (k + 1 < 27) gather(buf ^ 1, k + 1);
    // WMMA phase on current buffer
    {
      int m = wave * 16 + l16;
      const float* Ak = Wk + ((size_t)k * 64 + m) * C;
      const float* Bb = &Vt[buf][0];
      for (int c0 = 0; c0 < C; c0 += 4) {
        int ca = c0 + 2 * half;
        float2 aa = *(const float2*)&Ak[ca];
        float2 bb = *(const float2*)&Bb[l16 * 132 + ca];
        v2f a, b;
        a.x = aa.x; a.y = aa.y;
        b.x = bb.x; b.y = bb.y;
        acc = __builtin_amdgcn_wmma_f32_16x16x4_f32(false, a, false, b, (short)0, acc,
                                                    false, false);
      }
    }
    __syncthreads();
    buf ^= 1;
  }
  int col = nt * 16 + l16;
  if (col < N) {
    int rbase = wave * 16 + half * 8;
    for (int r = 0; r < 8; ++r) OUT[(rbase + r) * N + col] = acc[r];
  }
}

// ---------------------------------------------------------------------------
// 4) BatchNorm batch statistics: one block per channel (64 channels)
// ---------------------------------------------------------------------------
__global__ void k_bn_stats(const float* __restrict__ Y, int Nsp,
                           float* __restrict__ meanvar) {
  int ch = blockIdx.x;
  const float* y = Y + (size_t)ch * Nsp;
  float s = 0.f, s2 = 0.f;
  for (int i = threadIdx.x; i < Nsp; i += blockDim.x) {
    float v = y[i];
    s += v;
    s2 += v * v;
  }
  __shared__ float sh[256];
  __shared__ float sh2[256];
  sh[threadIdx.x] = s;
  sh2[threadIdx.x] = s2;
  __syncthreads();
  for (int off = 128; off > 0; off >>= 1) {
    if ((int)threadIdx.x < off) {
      sh[threadIdx.x] += sh[threadIdx.x + off];
      sh2[threadIdx.x] += sh2[threadIdx.x + off];
    }
    __syncthreads();
  }
  if (threadIdx.x == 0) {
    float mean = sh[0] / (float)Nsp;
    float var = sh2[0] / (float)Nsp - mean * mean;
    meanvar[ch] = mean;
    meanvar[64 + ch] = var;
  }
}

// ---------------------------------------------------------------------------
// 5) BN normalize + affine + ReLU
// ---------------------------------------------------------------------------
__global__ void k_bn_relu(const float* __restrict__ Y, const float* __restrict__ meanvar,
                          const float* __restrict__ gamma, const float* __restrict__ beta,
                          float* __restrict__ OUT, int Nsp) {
  int i = blockIdx.x * blockDim.x + threadIdx.x;
  if (i >= 64 * Nsp) return;
  int ch = i / Nsp;
  float mean = meanvar[ch];
  float var = meanvar[64 + ch];
  float xn = (Y[i] - mean) * rsqrtf(var + BN_EPS);
  float v = xn * gamma[ch] + beta[ch];
  OUT[i] = v > 0.f ? v : 0.f;
}

// ---------------------------------------------------------------------------
extern "C" void kernel_launch(void* const* d_in, const int* in_sizes, int n_in,
                              void* d_out, int out_size, void* d_ws, size_t ws_size,
                              hipStream_t stream) {
  const float* x1     = (const float*)d_in[0];   // [1,64,8,12,12]
  const float* x2     = (const float*)d_in[1];   // [1,64,16,24,24]
  const float* off_w1 = (const float*)d_in[2];   // [81,128,3,3,3]
  const float* off_b1 = (const float*)d_in[3];   // [81]
  const float* w1     = (const float*)d_in[4];   // [64,128,27]
  const float* g1     = (const float*)d_in[5];
  const float* b1     = (const float*)d_in[6];
  const float* off_w2 = (const float*)d_in[7];   // [81,64,3,3,3]
  const float* off_b2 = (const float*)d_in[8];
  const float* w2     = (const float*)d_in[9];   // [64,64,27]
  const float* g2     = (const float*)d_in[10];
  const float* b2     = (const float*)d_in[11];
  float* out = (float*)d_out;                    // [64,12,20,20] = 307200

  const int N1 = 14 * 22 * 22;   // 6776
  const int N2 = 12 * 20 * 20;   // 4800

  // workspace layout (floats)
  float* ws    = (float*)d_ws;
  float* X     = ws;                  // 128*9216 = 1179648
  float* off1  = X    + 1179648;      // 81*N1    = 548856
  float* y1    = off1 + 548856;       // 64*N1    = 433664
  float* a1    = y1   + 433664;       // 64*N1    = 433664
  float* off2  = a1   + 433664;       // 81*N2    = 388800
  float* y2    = off2 + 388800;       // 64*N2    = 307200
  float* stats = y2   + 307200;       // 256
  float* Wk1   = stats + 256;         // 27*64*128 = 221184
  float* Wk2   = Wk1  + 221184;       // 27*64*64  = 110592

  // 0) weight rearrangement for deform GEMM A-fragments
  {
    int n = 64 * 128 * 27;
    k_wt_to_kmc<<<(n + 255) / 256, 256, 0, stream>>>(w1, Wk1, 128);
    n = 64 * 64 * 27;
    k_wt_to_kmc<<<(n + 255) / 256, 256, 0, stream>>>(w2, Wk2, 64);
  }
  // 1) upsample + concat
  {
    int n = 128 * 16 * 24 * 24;
    k_upcat<<<(n + 255) / 256, 256, 0, stream>>>(x1, x2, X);
  }
  // 2) offset conv 1: M=81, C=128 -> off1[81,N1]
  {
    int blocks = (N1 + 15) / 16;
    k_conv_gemm_wmma<<<blocks, 192, 0, stream>>>(X, off_w1, off_b1, off1,
                                                 128, 16, 24, 24, 81, 14, 22, 22);
  }
  // 3) deform GEMM 1 -> y1[64,N1]
  {
    int blocks = (N1 + 15) / 16;
    k_deform_gemm_wmma<<<blocks, 128, 0, stream>>>(X, off1, Wk1, y1,
                                                   128, 16, 24, 24, 14, 22, 22);
  }
  // 4) BN stats + BN/ReLU -> a1
  k_bn_stats<<<64, 256, 0, stream>>>(y1, N1, stats);
  {
    int n = 64 * N1;
    k_bn_relu<<<(n + 255) / 256, 256, 0, stream>>>(y1, stats, g1, b1, a1, N1);
  }
  // 5) offset conv 2: M=81, C=64 -> off2[81,N2]
  {
    int blocks = (N2 + 15) / 16;
    k_conv_gemm_wmma<<<blocks, 192, 0, stream>>>(a1, off_w2, off_b2, off2,
                                                 64, 14, 22, 22, 81, 12, 20, 20);
  }
  // 6) deform GEMM 2 -> y2[64,N2]
  {
    int blocks = (N2 + 15) / 16;
    k_deform_gemm_wmma<<<blocks, 128, 0, stream>>>(a1, off2, Wk2, y2,
                                                   64, 14, 22, 22, 12, 20, 20);
  }
  // 7) BN stats + BN/ReLU -> out
  k_bn_stats<<<64, 256, 0, stream>>>(y2, N2, stats + 128);
  {
    int n = 64 * N2;
    k_bn_relu<<<(n + 255) / 256, 256, 0, stream>>>(y2, stats + 128, g2, b2, out, N2);
  }
}